// SRenderY_65798898975044
// MI455X (gfx1250) — compile-verified
//
#include <hip/hip_runtime.h>
#include <stdint.h>
#include <stddef.h>

namespace {
constexpr int kB  = 8;
constexpr int kV  = 5023;
constexpr int kF  = 9976;
constexpr int kH  = 512;
constexpr int kHP = kH * kH;       // pixels per image (262144)
constexpr int kPix = kB * kHP;     // total pixels (2097152)
}

typedef float v2f __attribute__((ext_vector_type(2)));

// ---------------------------------------------------------------------------
// Kernel 1: zero the normal accumulators in workspace
// ---------------------------------------------------------------------------
__global__ void knl_zero(float* __restrict__ p, int n) {
  int i = blockIdx.x * blockDim.x + threadIdx.x;
  if (i < n) p[i] = 0.0f;
}

// ---------------------------------------------------------------------------
// Kernel 2: per-face cross products, scatter-add into per-vertex accumulators
// ---------------------------------------------------------------------------
__device__ __forceinline__ void atomic_add3(float* base, float x, float y, float z) {
  atomicAdd(base + 0, x);
  atomicAdd(base + 1, y);
  atomicAdd(base + 2, z);
}

__device__ __forceinline__ void accum_vertex_normals(
    float x0, float y0, float z0,
    float x1, float y1, float z1,
    float x2, float y2, float z2,
    int i0, int i1, int i2,
    float* __restrict__ acc) {
  // n[i1] += cross(v2 - v1, v0 - v1)
  {
    float ax = x2 - x1, ay = y2 - y1, az = z2 - z1;
    float bx = x0 - x1, by = y0 - y1, bz = z0 - z1;
    atomic_add3(acc + 3 * i1, ay * bz - az * by, az * bx - ax * bz, ax * by - ay * bx);
  }
  // n[i2] += cross(v0 - v2, v1 - v2)
  {
    float ax = x0 - x2, ay = y0 - y2, az = z0 - z2;
    float bx = x1 - x2, by = y1 - y2, bz = z1 - z2;
    atomic_add3(acc + 3 * i2, ay * bz - az * by, az * bx - ax * bz, ax * by - ay * bx);
  }
  // n[i0] += cross(v1 - v0, v2 - v0)
  {
    float ax = x1 - x0, ay = y1 - y0, az = z1 - z0;
    float bx = x2 - x0, by = y2 - y0, bz = z2 - z0;
    atomic_add3(acc + 3 * i0, ay * bz - az * by, az * bx - ax * bz, ax * by - ay * bx);
  }
}

__global__ void __launch_bounds__(256)
knl_face_scatter(const float* __restrict__ verts,
                 const float* __restrict__ tverts,
                 const int*   __restrict__ faces,
                 float* __restrict__ nacc,
                 float* __restrict__ tnacc) {
  int idx = blockIdx.x * blockDim.x + threadIdx.x;
  if (idx >= kB * kF) return;
  int b = idx / kF;
  int f = idx - b * kF;
  int i0 = faces[3 * f + 0];
  int i1 = faces[3 * f + 1];
  int i2 = faces[3 * f + 2];
  size_t vb = (size_t)b * kV * 3;

  const float* v = verts + vb;
  accum_vertex_normals(v[3*i0], v[3*i0+1], v[3*i0+2],
                       v[3*i1], v[3*i1+1], v[3*i1+2],
                       v[3*i2], v[3*i2+1], v[3*i2+2],
                       i0, i1, i2, nacc + vb);

  // transformed vertices get z += 10 BEFORE differencing (faithful to ref fp)
  const float* tv = tverts + vb;
  accum_vertex_normals(tv[3*i0], tv[3*i0+1], tv[3*i0+2] + 10.0f,
                       tv[3*i1], tv[3*i1+1], tv[3*i1+2] + 10.0f,
                       tv[3*i2], tv[3*i2+1], tv[3*i2+2] + 10.0f,
                       i0, i1, i2, tnacc + vb);
}

// ---------------------------------------------------------------------------
// Kernel 3: normalize accumulators -> d_out normals / t_normals slices
// ---------------------------------------------------------------------------
__global__ void __launch_bounds__(256)
knl_normalize(const float* __restrict__ nacc,
              const float* __restrict__ tnacc,
              float* __restrict__ out_n,
              float* __restrict__ out_tn) {
  int i = blockIdx.x * blockDim.x + threadIdx.x;
  if (i >= kB * kV) return;
  {
    float x = nacc[3*i], y = nacc[3*i+1], z = nacc[3*i+2];
    float d = fmaxf(sqrtf(x*x + y*y + z*z), 1e-6f);
    out_n[3*i] = x / d; out_n[3*i+1] = y / d; out_n[3*i+2] = z / d;
  }
  {
    float x = tnacc[3*i], y = tnacc[3*i+1], z = tnacc[3*i+2];
    float d = fmaxf(sqrtf(x*x + y*y + z*z), 1e-6f);
    out_tn[3*i] = x / d; out_tn[3*i+1] = y / d; out_tn[3*i+2] = z / d;
  }
}

// ---------------------------------------------------------------------------
// Kernel 4: rasterize.
//  - Streaming operands (bary 3KB + p2f 1KB per block) staged cooperatively
//    into LDS with one global_load_async_to_lds_b128 per lane (ASYNCcnt path).
//  - Gathers (faces/uvcoords/normal tables, ~1.4MB total) stay L2-resident.
//  - Image-sized outputs written once with non-temporal stores.
// ---------------------------------------------------------------------------
__global__ void __launch_bounds__(256)
knl_rasterize(const int*   __restrict__ pix_to_face,
              const float* __restrict__ bary,
              const int*   __restrict__ faces,
              const float* __restrict__ uvc,
              const float* __restrict__ nrm,   // normalized vertex normals (B*V*3)
              const float* __restrict__ tnv,   // normalized t-normals (B*V*3)
              float* __restrict__ out_uv,      // (B,3,H,H)
              float* __restrict__ out_mask,    // (B,1,H,H)
              float* __restrict__ out_grid,    // (B,H,H,2)
              float* __restrict__ out_nimg) {  // (B,3,H,H)
  __shared__ __align__(16) float s_bary[256 * 3];   // 3072 B
  __shared__ __align__(16) int   s_p2f[256];        // 1024 B

  const int t    = threadIdx.x;
  const int pix0 = blockIdx.x * 256;                // kPix divisible by 256
  const int pix  = pix0 + t;

#if defined(__HIP_DEVICE_COMPILE__)
  {
    // Cooperative async DMA: lanes 0..191 stage the 3KB bary tile, lanes
    // 192..255 stage the 1KB p2f tile; 16B per lane, all bases 16B-aligned
    // (3072*blockIdx / 1024*blockIdx byte offsets). Address select is a
    // cndmask, so the async op issues with full EXEC.
    const bool isB = (t < 192);
    unsigned laddr = isB
        ? (unsigned)(uintptr_t)s_bary + (unsigned)(t * 16)
        : (unsigned)(uintptr_t)s_p2f  + (unsigned)((t - 192) * 16);
    const char* gaddr = isB
        ? (const char*)bary        + (size_t)pix0 * 12 + (size_t)t * 16
        : (const char*)pix_to_face + (size_t)pix0 * 4  + (size_t)(t - 192) * 16;
    asm volatile("global_load_async_to_lds_b128 %0, %1, off"
                 :: "v"(laddr), "v"(gaddr) : "memory");
    // Each wave drains its own ASYNCcnt; barrier publishes the LDS tile
    // across waves (staging lanes != consuming lanes).
    asm volatile("s_wait_asynccnt 0" ::: "memory");
  }
#else
  s_bary[3*t+0] = bary[(size_t)pix*3+0];
  s_bary[3*t+1] = bary[(size_t)pix*3+1];
  s_bary[3*t+2] = bary[(size_t)pix*3+2];
  s_p2f[t] = pix_to_face[pix];
#endif
  __syncthreads();

  const float w0 = s_bary[3*t+0];
  const float w1 = s_bary[3*t+1];
  const float w2 = s_bary[3*t+2];
  const int p2f  = s_p2f[t];

  float uv[3] = {0.0f, 0.0f, 0.0f};
  float nn[3] = {0.0f, 0.0f, 0.0f};
  float tnz = 0.0f;

  if (p2f >= 0) {
    // p2f indexes the flattened (B*F) face table: faces may come from any batch
    int bp = p2f / kF;
    int f  = p2f - bp * kF;
    __builtin_prefetch(&uvc[9 * f], 0, 0);      // global_prefetch_b8
    int i0 = faces[3 * f + 0];
    int i1 = faces[3 * f + 1];
    int i2 = faces[3 * f + 2];
    size_t g0 = ((size_t)bp * kV + i0) * 3;
    size_t g1 = ((size_t)bp * kV + i1) * 3;
    size_t g2 = ((size_t)bp * kV + i2) * 3;

    #pragma unroll
    for (int d = 0; d < 3; ++d) {
      uv[d] = w0 * (uvc[9*f     + d] * 0.5f + 0.5f)
            + w1 * (uvc[9*f + 3 + d] * 0.5f + 0.5f)
            + w2 * (uvc[9*f + 6 + d] * 0.5f + 0.5f);
      nn[d] = w0 * nrm[g0 + d] + w1 * nrm[g1 + d] + w2 * nrm[g2 + d];
    }
    // Only z of the interpolated t-normal survives into pos_mask.
    tnz = w0 * tnv[g0 + 2] + w1 * tnv[g1 + 2] + w2 * tnv[g2 + 2];
  }

  const int b   = pix / kHP;           // power-of-two divide
  const int rem = pix - b * kHP;       // y*H + x

  // Outputs: written once, never re-read -> non-temporal stores.
  __builtin_nontemporal_store(uv[0], &out_uv[(size_t)(b * 3 + 0) * kHP + rem]);
  __builtin_nontemporal_store(uv[1], &out_uv[(size_t)(b * 3 + 1) * kHP + rem]);
  __builtin_nontemporal_store(uv[2], &out_uv[(size_t)(b * 3 + 2) * kHP + rem]);

  __builtin_nontemporal_store((tnz < -0.05f) ? 1.0f : 0.0f,
                              &out_mask[(size_t)b * kHP + rem]);

  v2f g; g.x = uv[0]; g.y = uv[1];
  __builtin_nontemporal_store(g, (v2f*)&out_grid[((size_t)b * kHP + rem) * 2]);

  __builtin_nontemporal_store(nn[0], &out_nimg[(size_t)(b * 3 + 0) * kHP + rem]);
  __builtin_nontemporal_store(nn[1], &out_nimg[(size_t)(b * 3 + 1) * kHP + rem]);
  __builtin_nontemporal_store(nn[2], &out_nimg[(size_t)(b * 3 + 2) * kHP + rem]);
}

// ---------------------------------------------------------------------------
// Launch
// ---------------------------------------------------------------------------
extern "C" void kernel_launch(void* const* d_in, const int* in_sizes, int n_in,
                              void* d_out, int out_size, void* d_ws, size_t ws_size,
                              hipStream_t stream) {
  (void)in_sizes; (void)n_in; (void)out_size; (void)ws_size;

  const float* verts  = (const float*)d_in[0];  // (B,V,3)
  const float* tverts = (const float*)d_in[1];  // (B,V,3)
  // d_in[2] = albedos: dead in the reference
  const float* uvc    = (const float*)d_in[3];  // (F,3,3)
  const float* bary   = (const float*)d_in[4];  // (B,H,H,1,3)
  const int*   faces  = (const int*)d_in[5];    // (F,3)
  const int*   p2f    = (const int*)d_in[6];    // (B,H,H,1)

  float* out = (float*)d_out;
  // Output tuple layout (flat, return order):
  const size_t o_uv   = 0;                               // uvcoords_images (B,3,H,H)
  const size_t o_mask = o_uv   + (size_t)kB * 3 * kHP;   // pos_mask (B,1,H,H)
  const size_t o_grid = o_mask + (size_t)kB * kHP;       // grid (B,H,H,2)
  const size_t o_norm = o_grid + (size_t)kB * kHP * 2;   // normals (B,V,3)
  const size_t o_nimg = o_norm + (size_t)kB * kV * 3;    // normal_images (B,3,H,H)
  const size_t o_tn   = o_nimg + (size_t)kB * 3 * kHP;   // t_normals (B,V,3)

  float* nacc  = (float*)d_ws;                 // (B,V,3) accumulator
  float* tnacc = nacc + (size_t)kB * kV * 3;   // (B,V,3) accumulator

  const int n_acc = 2 * kB * kV * 3;
  knl_zero<<<(n_acc + 255) / 256, 256, 0, stream>>>(nacc, n_acc);

  knl_face_scatter<<<(kB * kF + 255) / 256, 256, 0, stream>>>(
      verts, tverts, faces, nacc, tnacc);

  knl_normalize<<<(kB * kV + 255) / 256, 256, 0, stream>>>(
      nacc, tnacc, out + o_norm, out + o_tn);

  knl_rasterize<<<kPix / 256, 256, 0, stream>>>(
      p2f, bary, faces, uvc, out + o_norm, out + o_tn,
      out + o_uv, out + o_mask, out + o_grid, out + o_nimg);
}